// FurthestPointsSample_56521769615777
// MI455X (gfx1250) — compile-verified
//
#include <hip/hip_runtime.h>
#include <stdint.h>

// Furthest Point Sampling for x:[B=16, 3, N=65536] -> out:[B, 3, NP=1024] (fp32).
// One workgroup per batch; 512 threads (16 wave32 waves, 4 waves/SIMD, ~256-VGPR
// budget so the 128 resident dist registers stay in VGPRs).
// - 24576 points (288KB) staged into LDS once via gfx1250 async global->LDS DMA.
// - All point traffic is float4 (b128) wide.
// - Global-region loop runs FIRST; sched_barrier mask 0x0087 (ALU+DS may cross,
//   VMEM may not) lets the LDS-chunk work fill the L2-load latency shadow.
// - Query point carried through LDS; fetched from LDS when the selected point
//   is in the staged slab (shortens the serial critical path).
// - Argmax via packed u64 (distBits<<32 | (N-1-i)) : wave32 shfl_xor + LDS tree.

#define FPS_B   16
#define FPS_N   65536
#define FPS_NP  1024
#define FPS_T   512                            // threads per block (16 waves)
#define FPS_LP  24576                          // points resident in LDS (288 KB)
#define C_LDS   (FPS_LP / 4 / FPS_T)           // 12 float4-chunks/thread from LDS
#define C_GLB   ((FPS_N - FPS_LP) / 4 / FPS_T) // 20 float4-chunks/thread from L2
#define NPT     (4 * (C_LDS + C_GLB))          // 128 points per thread
#define SMEM_BYTES (3 * FPS_LP * 4 + 32 * 8 + 16)

// 12 async 16B copies per plane; INST_OFFSET applies to BOTH LDS and global addr.
// Chunk stride = 512 threads * 16B = 8192 bytes.
#define STAGE_PLANE(ldsbase, gbase)                                              \
    do {                                                                         \
        uint32_t _l = (uint32_t)(ldsbase) + (uint32_t)t * 16u;                   \
        uint64_t _g = (uint64_t)(uintptr_t)(gbase) + (uint32_t)t * 16u;          \
        asm volatile("global_load_async_to_lds_b128 %0, %1, off"                 \
                     :: "v"(_l), "v"(_g) : "memory");                            \
        asm volatile("global_load_async_to_lds_b128 %0, %1, off offset:8192"     \
                     :: "v"(_l), "v"(_g) : "memory");                            \
        asm volatile("global_load_async_to_lds_b128 %0, %1, off offset:16384"    \
                     :: "v"(_l), "v"(_g) : "memory");                            \
        asm volatile("global_load_async_to_lds_b128 %0, %1, off offset:24576"    \
                     :: "v"(_l), "v"(_g) : "memory");                            \
        asm volatile("global_load_async_to_lds_b128 %0, %1, off offset:32768"    \
                     :: "v"(_l), "v"(_g) : "memory");                            \
        asm volatile("global_load_async_to_lds_b128 %0, %1, off offset:40960"    \
                     :: "v"(_l), "v"(_g) : "memory");                            \
        asm volatile("global_load_async_to_lds_b128 %0, %1, off offset:49152"    \
                     :: "v"(_l), "v"(_g) : "memory");                            \
        asm volatile("global_load_async_to_lds_b128 %0, %1, off offset:57344"    \
                     :: "v"(_l), "v"(_g) : "memory");                            \
        asm volatile("global_load_async_to_lds_b128 %0, %1, off offset:65536"    \
                     :: "v"(_l), "v"(_g) : "memory");                            \
        asm volatile("global_load_async_to_lds_b128 %0, %1, off offset:73728"    \
                     :: "v"(_l), "v"(_g) : "memory");                            \
        asm volatile("global_load_async_to_lds_b128 %0, %1, off offset:81920"    \
                     :: "v"(_l), "v"(_g) : "memory");                            \
        asm volatile("global_load_async_to_lds_b128 %0, %1, off offset:90112"    \
                     :: "v"(_l), "v"(_g) : "memory");                            \
    } while (0)

__global__ __launch_bounds__(FPS_T, 1)
void fps_kernel(const float* __restrict__ x, float* __restrict__ out) {
    extern __shared__ float smem[];
    float* lpx = smem;
    float* lpy = smem + FPS_LP;
    float* lpz = smem + 2 * FPS_LP;
    const float4* lpx4 = (const float4*)lpx;
    const float4* lpy4 = (const float4*)lpy;
    const float4* lpz4 = (const float4*)lpz;
    unsigned long long* wbest = (unsigned long long*)(smem + 3 * FPS_LP); // [16]
    float4* qs = (float4*)((char*)wbest + 32 * 8);                       // [1]

    const int b = blockIdx.x;
    const int t = threadIdx.x;

    const float* xs = x + (size_t)b * 3 * FPS_N;
    const float* ys = xs + FPS_N;
    const float* zs = ys + FPS_N;
    const float4* xs4 = (const float4*)xs;
    const float4* ys4 = (const float4*)ys;
    const float4* zs4 = (const float4*)zs;
    float* ob = out + (size_t)b * 3 * FPS_NP;

    // ---- Stage first FPS_LP points of each plane into LDS with async DMA ----
    STAGE_PLANE((uint32_t)(uintptr_t)lpx, xs);
    STAGE_PLANE((uint32_t)(uintptr_t)lpy, ys);
    STAGE_PLANE((uint32_t)(uintptr_t)lpz, zs);
    asm volatile("s_wait_asynccnt 0" ::: "memory");

    // First selection is point 0: thread 0 emits output + seeds the LDS query.
    if (t == 0) {
        float q0x = xs[0], q0y = ys[0], q0z = zs[0];
        ob[0]          = q0x;
        ob[FPS_NP]     = q0y;
        ob[2 * FPS_NP] = q0z;
        *qs = make_float4(q0x, q0y, q0z, 0.0f);
    }
    __syncthreads();

    // ---- Per-thread running min distances live in VGPRs ----
    // dist[0..4*C_GLB)   : global-region points (processed first)
    // dist[4*C_GLB..NPT) : LDS-staged points
    float dist[NPT];
    #pragma unroll
    for (int k = 0; k < NPT; ++k) dist[k] = 1e10f;

    for (int j = 1; j < FPS_NP; ++j) {
        const float4 q = *qs;   // one ds_load_b128 heads the iteration

        unsigned long long best = 0ull;

        auto upd = [&](float ex, float ey, float ez, int i, int k) {
            float dx = ex - q.x;
            float dy = ey - q.y;
            float dz = ez - q.z;
            float d2 = fmaf(dx, dx, fmaf(dy, dy, dz * dz));
            float dm = fminf(dist[k], d2);
            dist[k] = dm;
            unsigned long long key =
                ((unsigned long long)__float_as_uint(dm) << 32) |
                (unsigned)(FPS_N - 1 - i);
            best = key > best ? key : best;
        };

        // Global (L2-resident) region FIRST: 20 chunks x 3 planes of b128.
        // Barrier mask 0x0087: ALU and DS may cross (so LDS-chunk work can fill
        // the L2 latency shadow) but VMEM may not (bounds in-flight loads).
        #pragma unroll
        for (int c = 0; c < C_GLB; ++c) {
            int ch = FPS_LP / 4 + t + c * FPS_T;
            float4 px = xs4[ch];
            float4 py = ys4[ch];
            float4 pz = zs4[ch];
            int i0 = 4 * ch, k0 = 4 * c;
            upd(px.x, py.x, pz.x, i0 + 0, k0 + 0);
            upd(px.y, py.y, pz.y, i0 + 1, k0 + 1);
            upd(px.z, py.z, pz.z, i0 + 2, k0 + 2);
            upd(px.w, py.w, pz.w, i0 + 3, k0 + 3);
            if ((c & 3) == 3) __builtin_amdgcn_sched_barrier(0x0087);
        }
        // LDS-resident slab: 12 chunks x 3 planes of ds_load_b128
        #pragma unroll
        for (int c = 0; c < C_LDS; ++c) {
            int ch = t + c * FPS_T;
            float4 px = lpx4[ch];
            float4 py = lpy4[ch];
            float4 pz = lpz4[ch];
            int i0 = 4 * ch, k0 = 4 * (C_GLB + c);
            upd(px.x, py.x, pz.x, i0 + 0, k0 + 0);
            upd(px.y, py.y, pz.y, i0 + 1, k0 + 1);
            upd(px.z, py.z, pz.z, i0 + 2, k0 + 2);
            upd(px.w, py.w, pz.w, i0 + 3, k0 + 3);
            if ((c & 3) == 3) __builtin_amdgcn_sched_barrier(0x0007);
        }

        // Wave32 max-reduction of packed keys
        #pragma unroll
        for (int off = 16; off > 0; off >>= 1) {
            unsigned long long o = __shfl_xor(best, off, 32);
            best = o > best ? o : best;
        }
        if ((t & 31) == 0) wbest[t >> 5] = best;
        __syncthreads();

        // Cross-wave reduction by wave 0 (16 waves in the block)
        if (t < 32) {
            unsigned long long b2 = (t < FPS_T / 32) ? wbest[t] : 0ull;
            #pragma unroll
            for (int off = 16; off > 0; off >>= 1) {
                unsigned long long o = __shfl_xor(b2, off, 32);
                b2 = o > b2 ? o : b2;
            }
            if (t == 0) {
                int s = (int)(FPS_N - 1u - (unsigned)(b2 & 0xffffffffu));
                float sx, sy, sz;
                if (s < FPS_LP) {          // staged slab: low-latency LDS fetch
                    sx = lpx[s]; sy = lpy[s]; sz = lpz[s];
                } else {
                    sx = xs[s];  sy = ys[s];  sz = zs[s];
                }
                ob[j]              = sx;
                ob[FPS_NP + j]     = sy;
                ob[2 * FPS_NP + j] = sz;
                *qs = make_float4(sx, sy, sz, 0.0f);  // next query via LDS
            }
        }
        __syncthreads();
    }
}

extern "C" void kernel_launch(void* const* d_in, const int* in_sizes, int n_in,
                              void* d_out, int out_size, void* d_ws, size_t ws_size,
                              hipStream_t stream) {
    (void)in_sizes; (void)n_in; (void)d_ws; (void)ws_size; (void)out_size;
    const float* x = (const float*)d_in[0];
    float* out = (float*)d_out;
    fps_kernel<<<dim3(FPS_B), dim3(FPS_T), SMEM_BYTES, stream>>>(x, out);
}